// CrossGAT_13632226197552
// MI455X (gfx1250) — compile-verified
//
#include <hip/hip_runtime.h>
#include <math.h>

#define NP 16384
#define NS 8192
#define DP 256
#define DSS 128
#define DI 128

typedef __attribute__((ext_vector_type(2))) float v2f;
typedef __attribute__((ext_vector_type(8))) float v8f;

// ---------------------------------------------------------------------------
// GEMM: C[M,N] = A[M,K] @ W[K,N] + bias[N], f32 via V_WMMA_F32_16X16X4_F32.
// One wave per 16x16 C tile; grid sized so every wave has a full tile
// (EXEC all-ones requirement for WMMA). 4 waves (128 thr) per block.
// ---------------------------------------------------------------------------
__global__ __launch_bounds__(128) void gemm_f32_wmma(
    const float* __restrict__ A, const float* __restrict__ W,
    const float* __restrict__ bias, float* __restrict__ C,
    int M, int N, int K) {
  const int lane = threadIdx.x & 31;
  const int wave = threadIdx.x >> 5;
  const int tiles_n = N >> 4;
  const int tile = blockIdx.x * 4 + wave;
  const int tm = tile / tiles_n;
  const int tn = tile - tm * tiles_n;
  const int row0 = tm << 4;
  const int col0 = tn << 4;
  const int mr = lane & 15;
  const int khalf = (lane >> 4) << 1;  // lanes 0-15 -> K base 0, lanes 16-31 -> 2

  v8f acc = {};
  const float* arow = A + (size_t)(row0 + mr) * K + khalf;  // A[M=mr, K=khalf+..]
  const float* bcol = W + (size_t)khalf * N + (col0 + mr);  // W[K=khalf+.., N=mr]

  for (int kb = 0; kb < K; kb += 4) {
    const float2 av = *(const float2*)(arow + kb);  // A[., kb+khalf], A[., kb+khalf+1]
    v2f a; a.x = av.x; a.y = av.y;
    v2f b; b.x = bcol[kb * N]; b.y = bcol[kb * N + N];
    acc = __builtin_amdgcn_wmma_f32_16x16x4_f32(
        false, a, false, b, (short)0, acc, false, false);
  }

  const float bv = bias[col0 + mr];
  const int mbase = row0 + ((lane >> 4) << 3);  // lanes 16-31 hold M = v+8
  const int n = col0 + mr;
#pragma unroll
  for (int v = 0; v < 8; ++v) {
    C[(size_t)(mbase + v) * N + n] = acc[v] + bv;
  }
}

// ---------------------------------------------------------------------------
// Row sums of proj matrix: one wave per row (N = 128 -> 4 loads/lane).
// ---------------------------------------------------------------------------
__global__ __launch_bounds__(256) void rowsum_kernel(
    const float* __restrict__ X, float* __restrict__ rs, int M, int N) {
  const int w = (int)((blockIdx.x * blockDim.x + threadIdx.x) >> 5);
  const int lane = threadIdx.x & 31;
  if (w >= M) return;
  const float* row = X + (size_t)w * N;
  float s = 0.f;
  for (int c = lane; c < N; c += 32) s += row[c];
  for (int off = 16; off > 0; off >>= 1) s += __shfl_down(s, off, 32);
  if (lane == 0) rs[w] = s;
}

// ---------------------------------------------------------------------------
// Softmax stats over gathered scores: block 0 -> (max, sumexp) of
// rs_prot[prot_idx[i]], block 1 -> same for sub. stats = {mp, Zp, ms, Zs}.
// ---------------------------------------------------------------------------
__global__ __launch_bounds__(1024) void softmax_stats(
    const float* __restrict__ rs_p, const int* __restrict__ idx_p, int np,
    const float* __restrict__ rs_s, const int* __restrict__ idx_s, int ns,
    float* __restrict__ stats) {
  const float* rs = (blockIdx.x == 0) ? rs_p : rs_s;
  const int* idx = (blockIdx.x == 0) ? idx_p : idx_s;
  const int n = (blockIdx.x == 0) ? np : ns;
  __shared__ float red[32];
  __shared__ float red2[32];
  const int t = threadIdx.x;
  const int nw = (int)(blockDim.x >> 5);

  float m = -3.4e38f;
  for (int i = t; i < n; i += blockDim.x) m = fmaxf(m, rs[idx[i]]);
  for (int off = 16; off > 0; off >>= 1) m = fmaxf(m, __shfl_down(m, off, 32));
  if ((t & 31) == 0) red[t >> 5] = m;
  __syncthreads();
  if (t < 32) {
    float v = (t < nw) ? red[t] : -3.4e38f;
    for (int off = 16; off > 0; off >>= 1) v = fmaxf(v, __shfl_down(v, off, 32));
    if (t == 0) red[0] = v;
  }
  __syncthreads();
  m = red[0];

  float z = 0.f;
  for (int i = t; i < n; i += blockDim.x) z += expf(rs[idx[i]] - m);
  for (int off = 16; off > 0; off >>= 1) z += __shfl_down(z, off, 32);
  if ((t & 31) == 0) red2[t >> 5] = z;
  __syncthreads();
  if (t < 32) {
    float v = (t < nw) ? red2[t] : 0.f;
    for (int off = 16; off > 0; off >>= 1) v += __shfl_down(v, off, 32);
    if (t == 0) { stats[blockIdx.x * 2] = m; stats[blockIdx.x * 2 + 1] = v; }
  }
}

// ---------------------------------------------------------------------------
// partial[blk][c] = (1/Z) * sum_{rows of this block} exp(rs[idx[r]]-m) * proj[r][c]
// Fixed block partials -> deterministic (no float atomics).
// threads 0..127 own column c (coalesced proj reads), 2 rows in flight.
// ---------------------------------------------------------------------------
__global__ __launch_bounds__(256) void weighted_partial(
    const float* __restrict__ proj, const float* __restrict__ rs,
    const int* __restrict__ idx, const float* __restrict__ stats, int statoff,
    int n, int nblk, float* __restrict__ partial) {
  const int c = threadIdx.x & 127;
  const int half = threadIdx.x >> 7;
  const float m = stats[statoff];
  const float Z = stats[statoff + 1];
  const int rows_per_blk = n / nblk;
  const int r0 = blockIdx.x * rows_per_blk;
  float acc = 0.f;
  for (int r = r0 + half; r < r0 + rows_per_blk; r += 2) {
    const float w = expf(rs[idx[r]] - m);
    acc += w * proj[(size_t)r * DI + c];
  }
  __shared__ float sh[128];
  if (half) sh[c] = acc;
  __syncthreads();
  if (!half) partial[(size_t)blockIdx.x * DI + c] = (acc + sh[c]) / Z;
}

// blockIdx 0 -> w_p from part_p, blockIdx 1 -> w_s from part_s
__global__ __launch_bounds__(128) void reduce_partials(
    const float* __restrict__ part_p, const float* __restrict__ part_s,
    int nblk, float* __restrict__ w_p, float* __restrict__ w_s) {
  const float* part = (blockIdx.x == 0) ? part_p : part_s;
  float* out = (blockIdx.x == 0) ? w_p : w_s;
  const int c = threadIdx.x;
  float s = 0.f;
  for (int b = 0; b < nblk; ++b) s += part[(size_t)b * DI + c];
  out[c] = s;
}

// prot_delta[256] = w_s @ W_prot_out + b_prot_out (block 0)
// sub_delta[128]  = w_p @ W_sub_out  + b_sub_out  (block 1)
__global__ __launch_bounds__(256) void delta_kernel(
    const float* __restrict__ w_p, const float* __restrict__ w_s,
    const float* __restrict__ W_po, const float* __restrict__ b_po,
    const float* __restrict__ W_so, const float* __restrict__ b_so,
    float* __restrict__ prot_delta, float* __restrict__ sub_delta) {
  const int d = threadIdx.x;
  if (blockIdx.x == 0) {
    float s = b_po[d];
    for (int k = 0; k < DI; ++k) s += w_s[k] * W_po[k * DP + d];
    prot_delta[d] = s;
  } else if (d < DSS) {
    float s = b_so[d];
    for (int k = 0; k < DI; ++k) s += w_p[k] * W_so[k * DSS + d];
    sub_delta[d] = s;
  }
}

// ---------------------------------------------------------------------------
// Finalize: out = [prot_node + prot_delta | prot_idx | sub_node + sub_delta |
// sub_idx], float4/int4 vectorized (all segments 16B aligned).
// ---------------------------------------------------------------------------
__global__ __launch_bounds__(256) void finalize_kernel(
    const float4* __restrict__ pn, const int4* __restrict__ pidx,
    const float4* __restrict__ sn, const int4* __restrict__ sidx,
    const float* __restrict__ pd, const float* __restrict__ sd,
    float4* __restrict__ outA, int4* __restrict__ outPI,
    float4* __restrict__ outS, int4* __restrict__ outSI) {
  const size_t SEG_A = (size_t)NP * DP / 4;
  const size_t SEG_B = SEG_A + NP / 4;
  const size_t SEG_C = SEG_B + (size_t)NS * DSS / 4;
  const size_t SEG_T = SEG_C + NS / 4;
  const size_t q = (size_t)blockIdx.x * blockDim.x + threadIdx.x;
  if (q >= SEG_T) return;
  if (q < SEG_A) {
    float4 v = pn[q];
    const int d = (int)((q * 4) & (DP - 1));
    v.x += pd[d]; v.y += pd[d + 1]; v.z += pd[d + 2]; v.w += pd[d + 3];
    outA[q] = v;
  } else if (q < SEG_B) {
    outPI[q - SEG_A] = pidx[q - SEG_A];
  } else if (q < SEG_C) {
    const size_t r = q - SEG_B;
    float4 v = sn[r];
    const int d = (int)((r * 4) & (DSS - 1));
    v.x += sd[d]; v.y += sd[d + 1]; v.z += sd[d + 2]; v.w += sd[d + 3];
    outS[r] = v;
  } else {
    outSI[q - SEG_C] = sidx[q - SEG_C];
  }
}

extern "C" void kernel_launch(void* const* d_in, const int* in_sizes, int n_in,
                              void* d_out, int out_size, void* d_ws, size_t ws_size,
                              hipStream_t stream) {
  const float* prot_node   = (const float*)d_in[0];
  const float* sub_node    = (const float*)d_in[1];
  const float* W_prot_proj = (const float*)d_in[2];
  const float* b_prot_proj = (const float*)d_in[3];
  const float* W_sub_proj  = (const float*)d_in[4];
  const float* b_sub_proj  = (const float*)d_in[5];
  const float* W_prot_out  = (const float*)d_in[6];
  const float* b_prot_out  = (const float*)d_in[7];
  const float* W_sub_out   = (const float*)d_in[8];
  const float* b_sub_out   = (const float*)d_in[9];
  const int*   prot_idx    = (const int*)d_in[10];
  const int*   sub_idx     = (const int*)d_in[11];

  // workspace carve-up (all 256B aligned)
  char* ws = (char*)d_ws;
  float* prot_proj = (float*)ws; ws += (size_t)NP * DI * 4;   // 8 MB
  float* sub_proj  = (float*)ws; ws += (size_t)NS * DI * 4;   // 4 MB
  float* rs_prot   = (float*)ws; ws += (size_t)NP * 4;
  float* rs_sub    = (float*)ws; ws += (size_t)NS * 4;
  float* stats     = (float*)ws; ws += 256;                   // {mp, Zp, ms, Zs}
  const int NBLK = 64;
  float* part_p    = (float*)ws; ws += (size_t)NBLK * DI * 4;
  float* part_s    = (float*)ws; ws += (size_t)NBLK * DI * 4;
  float* w_p       = (float*)ws; ws += 256;
  float* w_s       = (float*)ws; ws += 256;
  float* prot_delta = (float*)ws; ws += 1024;
  float* sub_delta  = (float*)ws; ws += 512;

  float* out   = (float*)d_out;
  float* outA  = out;                                          // NP*DP f32
  int*   outPI = (int*)(out + (size_t)NP * DP);                // NP i32
  float* outS  = out + (size_t)NP * DP + NP;                   // NS*DSS f32
  int*   outSI = (int*)(out + (size_t)NP * DP + NP + (size_t)NS * DSS);

  // 1. projections via f32 WMMA (exact tile coverage -> EXEC all-ones)
  gemm_f32_wmma<<<(NP / 16) * (DI / 16) / 4, 128, 0, stream>>>(
      prot_node, W_prot_proj, b_prot_proj, prot_proj, NP, DI, DP);
  gemm_f32_wmma<<<(NS / 16) * (DI / 16) / 4, 128, 0, stream>>>(
      sub_node, W_sub_proj, b_sub_proj, sub_proj, NS, DI, DSS);

  // 2. row sums (one wave per row)
  rowsum_kernel<<<NP * 32 / 256, 256, 0, stream>>>(prot_proj, rs_prot, NP, DI);
  rowsum_kernel<<<NS * 32 / 256, 256, 0, stream>>>(sub_proj, rs_sub, NS, DI);

  // 3. softmax stats (rank-1 collapse: row/col softmaxes are single vectors)
  softmax_stats<<<2, 1024, 0, stream>>>(rs_prot, prot_idx, NP,
                                        rs_sub, sub_idx, NS, stats);

  // 4. w_p = softmax(ps)^T @ prot_proj ; w_s = softmax(ss)^T @ sub_proj
  weighted_partial<<<NBLK, 256, 0, stream>>>(prot_proj, rs_prot, prot_idx,
                                             stats, 0, NP, NBLK, part_p);
  weighted_partial<<<NBLK, 256, 0, stream>>>(sub_proj, rs_sub, sub_idx,
                                             stats, 2, NS, NBLK, part_s);
  reduce_partials<<<2, 128, 0, stream>>>(part_p, part_s, NBLK, w_p, w_s);

  // 5. output projections of the two broadcast vectors
  delta_kernel<<<2, 256, 0, stream>>>(w_p, w_s, W_prot_out, b_prot_out,
                                      W_sub_out, b_sub_out, prot_delta, sub_delta);

  // 6. broadcast-add + idx pass-through (float4/int4)
  const size_t total4 = (size_t)NP * DP / 4 + NP / 4 + (size_t)NS * DSS / 4 + NS / 4;
  finalize_kernel<<<(unsigned)((total4 + 255) / 256), 256, 0, stream>>>(
      (const float4*)prot_node, (const int4*)prot_idx,
      (const float4*)sub_node, (const int4*)sub_idx,
      prot_delta, sub_delta,
      (float4*)outA, (int4*)outPI, (float4*)outS, (int4*)outSI);
}